// SurfaceAwareTokenizer_49211735277864
// MI455X (gfx1250) — compile-verified
//
#include <hip/hip_runtime.h>
#include <hip/hip_bf16.h>
#include <math.h>

// Problem constants (from reference)
#define BB 8
#define NN 4096
#define CC 256
#define TARGET 1024
#define KNN 16
#define NUM_HIGH 614
#define NUM_UNI 410
#define M_REM (NN - NUM_HIGH) // 3482

typedef __attribute__((ext_vector_type(2))) float v2f;
typedef __attribute__((ext_vector_type(8))) float v8f;

// ---------------------------------------------------------------------------
// Kernel 1: curvature via WMMA Gram matrix.
// Grid: B*32 blocks, 128 threads (4 waves). Each wave owns 32 query rows.
// LDS: SoA points (x,y,z: 3*4096 floats) + per-wave 32x17 scratch.
// d2(i,j) = |q_i|^2 + |c_j|^2 - 2 * <q_i, c_j>, Gram via V_WMMA_F32_16X16X4_F32
// ---------------------------------------------------------------------------
__global__ __launch_bounds__(128) void sat_curvature_kernel(
    const float* __restrict__ xyz, float* __restrict__ curv)
{
    extern __shared__ float smem[];
    float* sx = smem;            // 4096
    float* sy = sx + NN;         // 4096
    float* sz = sy + NN;         // 4096
    float* scratch = sz + NN;    // 4 waves * 32*17

    const int b      = blockIdx.x >> 5;   // 32 blocks per batch
    const int blkInB = blockIdx.x & 31;
    const int tid    = threadIdx.x;
    const int wave   = tid >> 5;
    const int lane   = tid & 31;
    const int lm     = lane & 15;
    const bool hiHalf = (lane >= 16);

    // Load this batch's point cloud to LDS (SoA).
    const float* base = xyz + (size_t)b * NN * 3;
    for (int i = tid; i < NN; i += 128) {
        sx[i] = base[i * 3 + 0];
        sy[i] = base[i * 3 + 1];
        sz[i] = base[i * 3 + 2];
    }
    __syncthreads();

    const int rowBase = blkInB * 128 + wave * 32;
    const int myRow   = rowBase + lane;
    float* wscr = scratch + wave * (32 * 17);

    // A fragments (loop-invariant): lane holds row m = lane%16,
    // K components: lanes 0-15 -> (K0,K1)=(x,y); lanes 16-31 -> (K2,K3)=(z,0)
    const int qr0 = rowBase + lm;
    const int qr1 = rowBase + 16 + lm;
    v2f A0, A1;
    A0.x = hiHalf ? sz[qr0] : sx[qr0];
    A0.y = hiHalf ? 0.0f    : sy[qr0];
    A1.x = hiHalf ? sz[qr1] : sx[qr1];
    A1.y = hiHalf ? 0.0f    : sy[qr1];

    const float qn = sx[myRow]*sx[myRow] + sy[myRow]*sy[myRow] + sz[myRow]*sz[myRow];

    // Sorted (ascending) 16 smallest squared distances, self excluded.
    float knn[KNN];
#pragma unroll
    for (int t = 0; t < KNN; ++t) knn[t] = INFINITY;

    const int r0 = hiHalf ? 8 : 0;

    for (int chunk = 0; chunk < NN / 16; ++chunk) {
        const int gn = chunk * 16 + lm;
        const float cx = sx[gn], cy = sy[gn], cz = sz[gn];
        const float cn = cx*cx + cy*cy + cz*cz;

        // B fragment: lane holds column n = lane%16;
        // lanes 0-15 -> rows (K0,K1)=(x,y); lanes 16-31 -> (K2,K3)=(z,0)
        v2f Bf;
        Bf.x = hiHalf ? cz   : cx;
        Bf.y = hiHalf ? 0.0f : cy;

        v8f Cz = {};
        v8f D0 = __builtin_amdgcn_wmma_f32_16x16x4_f32(
            false, A0, false, Bf, (short)0, Cz, false, false);
        v8f D1 = __builtin_amdgcn_wmma_f32_16x16x4_f32(
            false, A1, false, Bf, (short)0, Cz, false, false);

        // Deposit (|c_j|^2 - 2G) into stride-17 scratch (conflict-free).
        // D element r of this lane is row (r0+r), col lm.
#pragma unroll
        for (int r = 0; r < 8; ++r) {
            wscr[(r0 + r) * 17 + lm]      = cn - 2.0f * D0[r];
            wscr[(16 + r0 + r) * 17 + lm] = cn - 2.0f * D1[r];
        }
        asm volatile("s_wait_dscnt 0" ::: "memory");

        // Lane owns scratch row `lane` (global row myRow): insert 16 candidates.
#pragma unroll
        for (int j = 0; j < 16; ++j) {
            const int col = chunk * 16 + j;
            float d2 = qn + wscr[lane * 17 + j];
            d2 = fmaxf(d2, 0.0f);
            if (col != myRow && d2 < knn[KNN - 1]) {
                knn[KNN - 1] = d2;
#pragma unroll
                for (int t = KNN - 1; t > 0; --t) {
                    const float lo = fminf(knn[t - 1], knn[t]);
                    const float hi = fmaxf(knn[t - 1], knn[t]);
                    knn[t - 1] = lo;
                    knn[t]     = hi;
                }
            }
        }
        asm volatile("s_wait_dscnt 0" ::: "memory");
    }

    // Unbiased variance of the 16 NN (Euclidean) distances.
    float sum = 0.0f;
    float d[KNN];
#pragma unroll
    for (int t = 0; t < KNN; ++t) { d[t] = sqrtf(knn[t]); sum += d[t]; }
    const float mean = sum * (1.0f / KNN);
    float var = 0.0f;
#pragma unroll
    for (int t = 0; t < KNN; ++t) { const float e = d[t] - mean; var += e * e; }
    var *= (1.0f / (KNN - 1));

    curv[b * NN + myRow] = var;
}

// ---------------------------------------------------------------------------
// Kernel 2: per batch, iterative argmax -> top-614 (descending, low-index ties),
// then wave-0 ballot compaction of the remaining 3482 indices (ascending).
// ---------------------------------------------------------------------------
__global__ __launch_bounds__(1024) void sat_topk_kernel(
    const float* __restrict__ curv, int* __restrict__ sel, int* __restrict__ rem)
{
    __shared__ float sv[NN];
    __shared__ float rv[32];
    __shared__ int   ri[32];

    const int b    = blockIdx.x;
    const int tid  = threadIdx.x;
    const int lane = tid & 31;
    const int wav  = tid >> 5;
    const float* c = curv + b * NN;

    for (int i = tid; i < NN; i += 1024) sv[i] = c[i];
    __syncthreads();

    for (int it = 0; it < NUM_HIGH; ++it) {
        float bv = -INFINITY; int bi = 0x7fffffff;
#pragma unroll
        for (int e = 0; e < NN / 1024; ++e) {
            const int i = tid + e * 1024;
            const float v = sv[i];
            if (v > bv || (v == bv && i < bi)) { bv = v; bi = i; }
        }
#pragma unroll
        for (int off = 16; off > 0; off >>= 1) {
            const float ov = __shfl_xor(bv, off, 32);
            const int   oi = __shfl_xor(bi, off, 32);
            if (ov > bv || (ov == bv && oi < bi)) { bv = ov; bi = oi; }
        }
        if (lane == 0) { rv[wav] = bv; ri[wav] = bi; }
        __syncthreads();
        if (wav == 0) {
            float v  = rv[lane];
            int   i2 = ri[lane];
#pragma unroll
            for (int off = 16; off > 0; off >>= 1) {
                const float ov = __shfl_xor(v, off, 32);
                const int   oi = __shfl_xor(i2, off, 32);
                if (ov > v || (ov == v && oi < i2)) { v = ov; i2 = oi; }
            }
            if (lane == 0) {
                sel[b * TARGET + it] = i2;
                sv[i2] = -INFINITY;
            }
        }
        __syncthreads();
    }

    // Ordered compaction of non-selected indices by wave 0.
    if (wav == 0) {
        int pos = 0;
        for (int basei = 0; basei < NN; basei += 32) {
            const int i = basei + lane;
            const bool keep = (sv[i] != -INFINITY);
            const unsigned mask = __builtin_amdgcn_ballot_w32(keep);
            const int off = __popc(mask & ((1u << lane) - 1u));
            if (keep) rem[b * M_REM + pos + off] = i;
            pos += __popc(mask);
        }
    }
}

// ---------------------------------------------------------------------------
// Kernel 3: farthest point sampling over the 3482 remaining points (410 picks).
// One block per batch, 512 threads; points + running min-dist live in LDS.
// ---------------------------------------------------------------------------
__global__ __launch_bounds__(512) void sat_fps_kernel(
    const float* __restrict__ xyz, const int* __restrict__ rem, int* __restrict__ sel)
{
    extern __shared__ float smem[];
    float* px = smem;             // 3482
    float* py = px + M_REM;
    float* pz = py + M_REM;
    float* dist = pz + M_REM;     // 3482
    __shared__ float rrv[16];
    __shared__ int   rri[16];
    __shared__ int   curSh;

    const int b    = blockIdx.x;
    const int tid  = threadIdx.x;
    const int lane = tid & 31;
    const int wav  = tid >> 5;
    const float* base = xyz + (size_t)b * NN * 3;
    const int* remb   = rem + b * M_REM;

    for (int i = tid; i < M_REM; i += 512) {
        const int gi = remb[i];
        px[i] = base[gi * 3 + 0];
        py[i] = base[gi * 3 + 1];
        pz[i] = base[gi * 3 + 2];
        dist[i] = INFINITY;
    }
    __syncthreads();

    int cur = 0;
    for (int it = 0; it < NUM_UNI; ++it) {
        if (tid == 0) sel[b * TARGET + NUM_HIGH + it] = remb[cur];
        const float cx = px[cur], cy = py[cur], cz = pz[cur];

        float bv = -INFINITY; int bi = 0x7fffffff;
        for (int i = tid; i < M_REM; i += 512) {
            const float dx = px[i] - cx, dy = py[i] - cy, dz = pz[i] - cz;
            const float dd = dx * dx + dy * dy + dz * dz;
            const float nd = fminf(dist[i], dd);
            dist[i] = nd;
            if (nd > bv || (nd == bv && i < bi)) { bv = nd; bi = i; }
        }
#pragma unroll
        for (int off = 16; off > 0; off >>= 1) {
            const float ov = __shfl_xor(bv, off, 32);
            const int   oi = __shfl_xor(bi, off, 32);
            if (ov > bv || (ov == bv && oi < bi)) { bv = ov; bi = oi; }
        }
        if (lane == 0) { rrv[wav] = bv; rri[wav] = bi; }
        __syncthreads();
        if (wav == 0) {
            float v  = (lane < 16) ? rrv[lane] : -INFINITY;
            int   i2 = (lane < 16) ? rri[lane] : 0x7fffffff;
#pragma unroll
            for (int off = 16; off > 0; off >>= 1) {
                const float ov = __shfl_xor(v, off, 32);
                const int   oi = __shfl_xor(i2, off, 32);
                if (ov > v || (ov == v && oi < i2)) { v = ov; i2 = oi; }
            }
            if (lane == 0) curSh = i2;
        }
        __syncthreads();
        cur = curSh;
    }
}

// ---------------------------------------------------------------------------
// Kernel 4: gather outputs. d_out = [xyz_out (B,1024,3) | feat_out (B,256,1024)]
// ---------------------------------------------------------------------------
__global__ __launch_bounds__(256) void sat_gather_kernel(
    const float* __restrict__ xyz, const float* __restrict__ feats,
    const int* __restrict__ sel, float* __restrict__ out)
{
    const int idx = blockIdx.x * 256 + threadIdx.x;
    const int XYZ_TOT = BB * TARGET * 3;
    const int FEAT_TOT = BB * CC * TARGET;
    if (idx < XYZ_TOT) {
        const int b  = idx / (TARGET * 3);
        const int r  = idx % (TARGET * 3);
        const int t  = r / 3;
        const int d  = r % 3;
        const int s  = sel[b * TARGET + t];
        out[idx] = xyz[((size_t)b * NN + s) * 3 + d];
    } else if (idx < XYZ_TOT + FEAT_TOT) {
        const int fi = idx - XYZ_TOT;
        const int b  = fi / (CC * TARGET);
        const int r  = fi % (CC * TARGET);
        const int c  = r / TARGET;
        const int t  = r % TARGET;
        const int s  = sel[b * TARGET + t];
        out[idx] = feats[((size_t)b * CC + c) * NN + s];
    }
}

// ---------------------------------------------------------------------------
// Launch
// ---------------------------------------------------------------------------
extern "C" void kernel_launch(void* const* d_in, const int* in_sizes, int n_in,
                              void* d_out, int out_size, void* d_ws, size_t ws_size,
                              hipStream_t stream) {
    const float* xyz   = (const float*)d_in[0];  // (B, N, 3) f32
    const float* feats = (const float*)d_in[1];  // (B, C, N) f32
    float* out = (float*)d_out;

    // Workspace layout (~269 KB total, fully rewritten every call):
    char* ws = (char*)d_ws;
    float* curv = (float*)ws;                                   // B*N f32
    int*   sel  = (int*)(ws + (size_t)BB * NN * 4);             // B*1024 i32
    int*   rem  = (int*)(ws + (size_t)BB * NN * 4 + (size_t)BB * TARGET * 4); // B*3482 i32

    // Phase 1: curvature (WMMA Gram-matrix kNN). 32 blocks/batch, 4 waves each.
    const size_t sh1 = (size_t)(3 * NN + 4 * 32 * 17) * sizeof(float);
    sat_curvature_kernel<<<dim3(BB * 32), dim3(128), sh1, stream>>>(xyz, curv);

    // Phase 2: top-614 + remainder compaction.
    sat_topk_kernel<<<dim3(BB), dim3(1024), 0, stream>>>(curv, sel, rem);

    // Phase 3: FPS over remainder.
    const size_t sh3 = (size_t)(4 * M_REM) * sizeof(float);
    sat_fps_kernel<<<dim3(BB), dim3(512), sh3, stream>>>(xyz, rem, sel);

    // Phase 4: gather outputs.
    const int total = BB * TARGET * 3 + BB * CC * TARGET;
    sat_gather_kernel<<<dim3((total + 255) / 256), dim3(256), 0, stream>>>(
        xyz, feats, sel, out);
}